// GCNcustom_12077448036415
// MI455X (gfx1250) — compile-verified
//
#include <hip/hip_runtime.h>
#include <hip/hip_bf16.h>

typedef float v2f __attribute__((ext_vector_type(2)));
typedef float v8f __attribute__((ext_vector_type(8)));

// ---------------------------------------------------------------------------
// Zero-fill (grid-stride) — d_ws is poisoned, so we must clear accumulators
// ourselves every call (deterministic, no cross-call state).
// ---------------------------------------------------------------------------
__global__ void sage_zero_kernel(float* __restrict__ p, long long n) {
  long long i = (long long)blockIdx.x * blockDim.x + threadIdx.x;
  long long stride = (long long)gridDim.x * blockDim.x;
  for (; i < n; i += stride) p[i] = 0.0f;
}

// ---------------------------------------------------------------------------
// Degree: cnt[dst[e]] += 1.0f  (f32 atomics resolve in L2; cnt is 200 KB)
// ---------------------------------------------------------------------------
__global__ void sage_degree_kernel(const long long* __restrict__ dst,
                                   float* __restrict__ cnt, int nedges) {
  int e = blockIdx.x * blockDim.x + threadIdx.x;
  if (e >= nedges) return;
  atomicAdd(cnt + dst[e], 1.0f);
}

// ---------------------------------------------------------------------------
// Scatter-add of gathered features: ssum[dst] += h[src]
// One thread per (edge, 4-feature chunk); consecutive threads read a
// contiguous 384B feature row (coalesced b128 loads), atomics land in L2
// (ssum is 19.2 MB << 192 MB L2).
// ---------------------------------------------------------------------------
__global__ void sage_scatter_kernel(const float* __restrict__ h,
                                    const long long* __restrict__ src,
                                    const long long* __restrict__ dst,
                                    float* __restrict__ ssum,
                                    int nedges, int f) {
  const int chunks = f >> 2;  // float4 chunks per feature row
  long long tid = (long long)blockIdx.x * blockDim.x + threadIdx.x;
  long long total = (long long)nedges * chunks;
  if (tid >= total) return;
  int e = (int)(tid / chunks);
  int c = (int)(tid % chunks) << 2;
  long long s = src[e];
  long long d = dst[e];
  const float4 v = *(const float4*)(h + s * (long long)f + c);
  float* o = ssum + d * (long long)f + c;
  atomicAdd(o + 0, v.x);
  atomicAdd(o + 1, v.y);
  atomicAdd(o + 2, v.z);
  atomicAdd(o + 3, v.w);
}

// ---------------------------------------------------------------------------
// Fused SAGE layer GEMM:  out = ReLU( (ssum/deg) @ Wl + h @ Wr + b )
// One wave32 per 16x16 output tile, V_WMMA_F32_16X16X4_F32, K-step 4.
//
// Operand layouts per CDNA5 ISA §7.12.2 (wave32):
//   A 16x4  f32: lane L, comp v -> A[L&15][k0 + v + 2*(L>>4)]   (float2 load)
//   B 4x16  f32: lane L, comp v -> B[k0 + v + 2*(L>>4)][L&15]
//   C/D 16x16  : lane L, comp v -> row (v + 8*(L>>4)), col (L&15)
// Guard is wave-uniform -> EXEC all-ones for every WMMA (ISA restriction).
// ---------------------------------------------------------------------------
__global__ void sage_gemm_relu_kernel(const float* __restrict__ ssum,
                                      const float* __restrict__ cnt,
                                      const float* __restrict__ h,
                                      const float* __restrict__ Wl,
                                      const float* __restrict__ Wr,
                                      const float* __restrict__ bias,
                                      float* __restrict__ out,
                                      int nnodes, int fin, int fout) {
  const int lane = threadIdx.x & 31;
  const int wave = blockIdx.x * (blockDim.x >> 5) + (threadIdx.x >> 5);
  const int ntile_n = fout >> 4;
  const int row_tile = wave / ntile_n;
  const int col_tile = wave - row_tile * ntile_n;
  if (row_tile * 16 >= nnodes) return;  // wave-uniform

  const int hi = lane >> 4;   // 0 or 1
  const int lo = lane & 15;
  const int row0 = row_tile * 16;
  const int col0 = col_tile * 16;

  // Mean normalization factor for this lane's A-row.
  const float deg = cnt[row0 + lo];
  const float inv = 1.0f / fmaxf(deg, 1.0f);

  const long long arow_base = (long long)(row0 + lo) * fin;
  v8f acc = {};

  for (int k0 = 0; k0 < fin; k0 += 4) {
    const int ka = k0 + 2 * hi;  // this half-wave's K pair

    // A tiles (contiguous float2 per lane)
    const float2 sm = *(const float2*)(ssum + arow_base + ka);
    const float2 sh = *(const float2*)(h + arow_base + ka);
    v2f a_mean, a_self;
    a_mean[0] = sm.x * inv;  a_mean[1] = sm.y * inv;
    a_self[0] = sh.x;        a_self[1] = sh.y;

    // B tiles: rows ka and ka+1 of the weight matrices, column col0+lo
    v2f bl, br;
    bl[0] = Wl[(long long)ka * fout + col0 + lo];
    bl[1] = Wl[(long long)(ka + 1) * fout + col0 + lo];
    br[0] = Wr[(long long)ka * fout + col0 + lo];
    br[1] = Wr[(long long)(ka + 1) * fout + col0 + lo];

    acc = __builtin_amdgcn_wmma_f32_16x16x4_f32(
        false, a_mean, false, bl, (short)0, acc, false, false);
    acc = __builtin_amdgcn_wmma_f32_16x16x4_f32(
        false, a_self, false, br, (short)0, acc, false, false);
  }

  // Epilogue: bias + ReLU, store D (lane L, comp v -> row v+8*hi, col lo)
  const float bn = bias[col0 + lo];
#pragma unroll
  for (int v = 0; v < 8; ++v) {
    const int m = v + 8 * hi;
    float val = acc[v] + bn;
    val = fmaxf(val, 0.0f);
    out[(long long)(row0 + m) * fout + col0 + lo] = val;
  }
}

// ---------------------------------------------------------------------------
// Host-side orchestration (graph-capture safe: only kernel launches).
// ---------------------------------------------------------------------------
static inline void launch_layer(const float* hin, const long long* src,
                                const long long* dst, const float* Wl,
                                const float* Wr, const float* b, float* ssum,
                                const float* cnt, float* hout, int nnodes,
                                int nedges, int fin, int fout,
                                hipStream_t stream) {
  // zero ssum
  {
    long long n = (long long)nnodes * fin;
    int blocks = 1024;
    sage_zero_kernel<<<blocks, 256, 0, stream>>>(ssum, n);
  }
  // scatter-add gathered features
  {
    long long total = (long long)nedges * (fin >> 2);
    int blocks = (int)((total + 255) / 256);
    sage_scatter_kernel<<<blocks, 256, 0, stream>>>(hin, src, dst, ssum,
                                                    nedges, fin);
  }
  // fused mean-normalize + dual GEMM + bias + ReLU
  {
    int waves = (nnodes / 16) * (fout / 16);
    int waves_per_block = 8;  // 256 threads = 8 wave32
    int blocks = (waves + waves_per_block - 1) / waves_per_block;
    sage_gemm_relu_kernel<<<blocks, waves_per_block * 32, 0, stream>>>(
        ssum, cnt, hin, Wl, Wr, b, hout, nnodes, fin, fout);
  }
}

extern "C" void kernel_launch(void* const* d_in, const int* in_sizes, int n_in,
                              void* d_out, int out_size, void* d_ws,
                              size_t ws_size, hipStream_t stream) {
  const int FIN = 96;   // FEATS[0..2]
  const int FOUT = 48;  // FEATS[3]

  const float* x = (const float*)d_in[0];
  const long long* ei = (const long long*)d_in[1];  // int64 per reference
  const float* Wl0 = (const float*)d_in[2];
  const float* Wr0 = (const float*)d_in[3];
  const float* b0 = (const float*)d_in[4];
  const float* Wl1 = (const float*)d_in[5];
  const float* Wr1 = (const float*)d_in[6];
  const float* b1 = (const float*)d_in[7];
  const float* Wl2 = (const float*)d_in[8];
  const float* Wr2 = (const float*)d_in[9];
  const float* b2 = (const float*)d_in[10];

  const int nnodes = in_sizes[0] / FIN;  // 50000
  const int nedges = in_sizes[1] / 2;    // 800000
  const long long* src = ei;
  const long long* dst = ei + nedges;

  // Workspace layout (floats): cnt[N] | ssum[N*96] | h1[N*96] | h2[N*96]
  float* cnt = (float*)d_ws;
  float* ssum = cnt + nnodes;
  float* h1 = ssum + (long long)nnodes * FIN;
  float* h2 = h1 + (long long)nnodes * FIN;
  float* outp = (float*)d_out;

  // Degree counts (recomputed every call; no cross-call state).
  sage_zero_kernel<<<256, 256, 0, stream>>>(cnt, nnodes);
  sage_degree_kernel<<<(nedges + 255) / 256, 256, 0, stream>>>(dst, cnt,
                                                               nedges);

  // Layer 0: 96 -> 96
  launch_layer(x, src, dst, Wl0, Wr0, b0, ssum, cnt, h1, nnodes, nedges, FIN,
               FIN, stream);
  // Layer 1: 96 -> 96
  launch_layer(h1, src, dst, Wl1, Wr1, b1, ssum, cnt, h2, nnodes, nedges, FIN,
               FIN, stream);
  // Layer 2: 96 -> 48 (writes final output)
  launch_layer(h2, src, dst, Wl2, Wr2, b2, ssum, cnt, outp, nnodes, nedges,
               FIN, FOUT, stream);
}